// DarkScratchDetectorLoss_9148280340496
// MI455X (gfx1250) — compile-verified
//
#include <hip/hip_runtime.h>
#include <hip/hip_bf16.h>
#include <math.h>

// Problem constants (match reference setup_inputs)
#define B_     32
#define P_     65536
#define NOBJ   16
#define CLS    4
#define TPB    256
#define CHUNKS (P_ / TPB)    // 256 blocks per batch row
#define NB3    (B_ * CHUNKS) // 8192 partial slots
#define POSBIT 0x100u        // matchinfo: bit8 = positive, bits[7:0] = object id

typedef __attribute__((ext_vector_type(2))) float v2f;
typedef __attribute__((ext_vector_type(8))) float v8f;

// ---------------------------------------------------------------------------
// Wave-level f32 sum on the matrix engine:
//   A = ones (16x4), B carries the 32 lane values in b.x (b.y = 0).
//   D[m,n] = sum_k B[k,n] (column sums, replicated over m); each lane's d[0]
//   is one of 16 distinct column sums -> 4 xor-shuffles finish the tree.
// EXEC must be all ones at the call site (callers use full 256-thread blocks
// with no divergence before the call).
// ---------------------------------------------------------------------------
__device__ __forceinline__ float wave_sum_wmma(float v) {
  v2f a; a[0] = 1.0f; a[1] = 1.0f;
  v2f b; b[0] = v;    b[1] = 0.0f;
  v8f c = {};
  c = __builtin_amdgcn_wmma_f32_16x16x4_f32(false, a, false, b, (short)0, c,
                                            false, false);
  float s = c[0];
  s += __shfl_xor(s, 1, 32);
  s += __shfl_xor(s, 2, 32);
  s += __shfl_xor(s, 4, 32);
  s += __shfl_xor(s, 8, 32);
  return s; // total across all 32 lanes, valid in every lane
}

// ---------------------------------------------------------------------------
// Kernel 1: per-prior IoU vs the 16 GT boxes. Writes packed matchinfo
// (obj id + pos flag) and the per-object argmax over priors via packed u64
// atomicMax (order-independent => deterministic; ties -> smallest prior).
// ---------------------------------------------------------------------------
__global__ void k_match(const float* __restrict__ priors_cxcy,
                        const float* __restrict__ boxes,
                        unsigned* __restrict__ matchinfo,
                        unsigned long long* __restrict__ objbest) {
  __shared__ float sbox[NOBJ * 4];
  __shared__ unsigned long long sbest[NOBJ];
  const int b = blockIdx.y;
  const int p = blockIdx.x * TPB + threadIdx.x;

  if (threadIdx.x < NOBJ * 4) sbox[threadIdx.x] = boxes[b * NOBJ * 4 + threadIdx.x];
  if (threadIdx.x < NOBJ) sbest[threadIdx.x] = 0ull;
  __syncthreads();

  const float4 pc = ((const float4*)priors_cxcy)[p];
  const float px0 = pc.x - pc.z * 0.5f, py0 = pc.y - pc.w * 0.5f;
  const float px1 = pc.x + pc.z * 0.5f, py1 = pc.y + pc.w * 0.5f;
  const float parea = (px1 - px0) * (py1 - py0);

  float best = -1.0f;
  int bestj = 0;
#pragma unroll
  for (int j = 0; j < NOBJ; ++j) {
    const float bx0 = sbox[j * 4 + 0], by0 = sbox[j * 4 + 1];
    const float bx1 = sbox[j * 4 + 2], by1 = sbox[j * 4 + 3];
    const float ix0 = fmaxf(bx0, px0), iy0 = fmaxf(by0, py0);
    const float ix1 = fminf(bx1, px1), iy1 = fminf(by1, py1);
    const float iw = fmaxf(ix1 - ix0, 0.0f), ih = fmaxf(iy1 - iy0, 0.0f);
    const float inter = iw * ih;
    const float barea = (bx1 - bx0) * (by1 - by0);
    const float iou = inter / fmaxf(parea + barea - inter, 1e-7f);
    if (iou > best) { best = iou; bestj = j; }  // first-max tie-break (argmax)
    // key: bigger IoU wins; equal IoU -> smaller prior index wins
    const unsigned long long key =
        ((unsigned long long)__float_as_uint(iou) << 32) |
        (unsigned long long)(0xFFFFFFFFu - (unsigned)p);
    atomicMax(&sbest[j], key);
  }
  // label_p = 0 iff ov < THRESHOLD-0.1 (= 0.4f under JAX weak-typed compare)
  matchinfo[b * P_ + p] = (unsigned)bestj | (best >= 0.4f ? POSBIT : 0u);
  __syncthreads();
  if (threadIdx.x < NOBJ)
    atomicMax(&objbest[b * NOBJ + threadIdx.x], sbest[threadIdx.x]);
}

// ---------------------------------------------------------------------------
// Kernel 2: forced-match fixup (sequential j loop => "last j wins" exactly).
// Forced priors become positive with ov = 1.0 (>= 0.4), so just set POSBIT.
// ---------------------------------------------------------------------------
__global__ void k_force(const unsigned long long* __restrict__ objbest,
                        unsigned* __restrict__ matchinfo) {
  const int b = blockIdx.x;
  if (threadIdx.x != 0) return;
  for (int j = 0; j < NOBJ; ++j) {
    const unsigned long long key = objbest[b * NOBJ + j];
    const unsigned iou_bits = (unsigned)(key >> 32);
    if (iou_bits != 0u) { // ov_obj > 0 (valid)
      const unsigned p = 0xFFFFFFFFu - (unsigned)(key & 0xFFFFFFFFull);
      matchinfo[b * P_ + p] = (unsigned)j | POSBIT;
    }
  }
}

// ---------------------------------------------------------------------------
// Kernel 3: decode + DIoU + label-smoothed CE, store conf_neg, per-block
// partial sums via WMMA wave reduction (deterministic, no float atomics).
// ---------------------------------------------------------------------------
__global__ void k_perprior(const float* __restrict__ odm_locs,
                           const float* __restrict__ odm_scores,
                           const float* __restrict__ priors_cxcy,
                           const float* __restrict__ boxes,
                           const int* __restrict__ labels,
                           const unsigned* __restrict__ matchinfo,
                           float* __restrict__ conf_neg,
                           float* __restrict__ locP,
                           float* __restrict__ confP,
                           int* __restrict__ nposP) {
  __shared__ float sL[TPB / 32], sC[TPB / 32];
  __shared__ int sN[TPB / 32];
  const int b = blockIdx.y;
  const int p = blockIdx.x * TPB + threadIdx.x;
  const int idx = b * P_ + p;

  const float4 g  = ((const float4*)odm_locs)[idx];
  const float4 pr = ((const float4*)priors_cxcy)[p];
  const float4 s  = ((const float4*)odm_scores)[idx];
  const unsigned mi = matchinfo[idx];
  const int obj   = (int)(mi & 0xFFu);
  const bool pos  = (mi & POSBIT) != 0u;
  const int lab   = labels[b * NOBJ + obj]; // always >= 1

  // SSD decode (variances 10, 5), cxcywh -> corners
  const float cx = g.x * pr.z * 0.1f + pr.x;
  const float cy = g.y * pr.w * 0.1f + pr.y;
  const float w  = expf(g.z * 0.2f) * pr.z;
  const float h  = expf(g.w * 0.2f) * pr.w;
  const float dx0 = cx - w * 0.5f, dy0 = cy - h * 0.5f;
  const float dx1 = cx + w * 0.5f, dy1 = cy + h * 0.5f;

  const float* tb = &boxes[(b * NOBJ + obj) * 4];
  const float tx0 = tb[0], ty0 = tb[1], tx1 = tb[2], ty1 = tb[3];

  // DIoU
  const float ix0 = fmaxf(dx0, tx0), iy0 = fmaxf(dy0, ty0);
  const float ix1 = fminf(dx1, tx1), iy1 = fminf(dy1, ty1);
  const float iw = fmaxf(ix1 - ix0, 0.0f), ih = fmaxf(iy1 - iy0, 0.0f);
  const float inter = iw * ih;
  const float ap = (dx1 - dx0) * (dy1 - dy0);
  const float at = (tx1 - tx0) * (ty1 - ty0);
  const float iou = inter / fmaxf(ap + at - inter, 1e-7f);
  const float cpx = (dx0 + dx1) * 0.5f, cpy = (dy0 + dy1) * 0.5f;
  const float ctx = (tx0 + tx1) * 0.5f, cty = (ty0 + ty1) * 0.5f;
  const float rho2 = (cpx - ctx) * (cpx - ctx) + (cpy - cty) * (cpy - cty);
  const float ex0 = fminf(dx0, tx0), ey0 = fminf(dy0, ty0);
  const float ex1 = fmaxf(dx1, tx1), ey1 = fmaxf(dy1, ty1);
  const float c2 = (ex1 - ex0) * (ex1 - ex0) + (ey1 - ey0) * (ey1 - ey0) + 1e-7f;
  const float di = fminf(fmaxf(iou - rho2 / c2, -1.0f), 1.0f);
  const float dloss = 1.0f - di;

  // label-smoothed CE over 4 classes
  const float m = fmaxf(fmaxf(s.x, s.y), fmaxf(s.z, s.w));
  const float e0 = expf(s.x - m), e1 = expf(s.y - m);
  const float e2 = expf(s.z - m), e3 = expf(s.w - m);
  const float lse = m + logf(e0 + e1 + e2 + e3);
  const float sum_logp = (s.x + s.y + s.z + s.w) - 4.0f * lse;
  const float sl[4] = {s.x, s.y, s.z, s.w};
  const int lbl = pos ? lab : 0;
  const float logp_l = sl[lbl] - lse;
  const float smooth = 0.05f / 3.0f;
  const float confw = 0.95f - smooth;
  const float conf_all = -(smooth * sum_logp + confw * logp_l);

  conf_neg[idx] = pos ? 0.0f : conf_all;

  // block-level partials (WMMA wave reduce, then tiny LDS reduce; EXEC all 1s)
  const float wl = wave_sum_wmma(pos ? dloss : 0.0f);
  const float wc = wave_sum_wmma(pos ? conf_all : 0.0f);
  const int wn = __popcll(__ballot(pos));
  const int wave = threadIdx.x >> 5, lane = threadIdx.x & 31;
  if (lane == 0) { sL[wave] = wl; sC[wave] = wc; sN[wave] = wn; }
  __syncthreads();
  if (threadIdx.x == 0) {
    float tl = 0.0f, tc = 0.0f; int tn = 0;
    for (int i = 0; i < TPB / 32; ++i) { tl += sL[i]; tc += sC[i]; tn += sN[i]; }
    const int pidx = b * CHUNKS + blockIdx.x;
    locP[pidx] = tl; confP[pidx] = tc; nposP[pidx] = tn;
  }
}

// ---------------------------------------------------------------------------
// Kernel 4: per-row n_pos, init radix-select state, zero bins.
// ---------------------------------------------------------------------------
__global__ void k_initrows(const int* __restrict__ nposP,
                           int* __restrict__ npos,
                           unsigned* __restrict__ prefix,
                           unsigned* __restrict__ krem,
                           unsigned* __restrict__ bins) {
  __shared__ int sn[TPB];
  const int b = blockIdx.x, t = threadIdx.x;
  sn[t] = nposP[b * CHUNKS + t]; // CHUNKS == TPB == 256
  bins[b * 256 + t] = 0u;
  __syncthreads();
  for (int off = TPB / 2; off > 0; off >>= 1) {
    if (t < off) sn[t] += sn[t + off];
    __syncthreads();
  }
  if (t == 0) {
    const int np = sn[0];
    npos[b] = np;
    long long K = 3LL * np;
    if (K > P_) K = P_;
    krem[b] = (unsigned)K;
    prefix[b] = 0u;
  }
}

// ---------------------------------------------------------------------------
// Kernel 5: radix-select histogram pass (count-only => deterministic).
// conf_neg >= 0 so float bits are monotone as u32.
// ---------------------------------------------------------------------------
__global__ void k_hist(const float* __restrict__ conf_neg,
                       const unsigned* __restrict__ prefix,
                       const unsigned* __restrict__ krem,
                       unsigned* __restrict__ bins, int pass) {
  __shared__ unsigned sb[256];
  const int b = blockIdx.y;
  sb[threadIdx.x] = 0u;
  __syncthreads();
  if (krem[b] != 0u) {
    const int p = blockIdx.x * TPB + threadIdx.x;
    // prefetch the chunk one block-stride ahead (same row, speculative)
    __builtin_prefetch(&conf_neg[b * P_ + ((p + TPB) & (P_ - 1))], 0, 1);
    const unsigned bits = __float_as_uint(conf_neg[b * P_ + p]);
    const int shift = 24 - 8 * pass;
    bool take = true;
    if (pass > 0) {
      const unsigned pr = prefix[b];
      take = ((bits >> (shift + 8)) == (pr >> (shift + 8)));
    }
    if (take) atomicAdd(&sb[(bits >> shift) & 0xFFu], 1u);
  }
  __syncthreads();
  if (sb[threadIdx.x]) atomicAdd(&bins[b * 256 + threadIdx.x], sb[threadIdx.x]);
}

// ---------------------------------------------------------------------------
// Kernel 6: pick the bin containing the K-th largest; update prefix/krem;
// zero bins for the next pass.
// ---------------------------------------------------------------------------
__global__ void k_scan(unsigned* __restrict__ bins,
                       unsigned* __restrict__ prefix,
                       unsigned* __restrict__ krem, int pass) {
  __shared__ unsigned sc[256];
  const int b = blockIdx.x, t = threadIdx.x;
  sc[t] = bins[b * 256 + t];
  bins[b * 256 + t] = 0u;
  __syncthreads();
  if (t == 0) {
    const unsigned kr = krem[b];
    if (kr != 0u) {
      const int shift = 24 - 8 * pass;
      unsigned cum = 0;
      int beta = 0;
      for (int bin = 255; bin >= 0; --bin) {
        const unsigned c = sc[bin];
        if (cum + c >= kr) { beta = bin; break; }
        cum += c;
      }
      prefix[b] |= ((unsigned)beta) << shift;
      krem[b] = kr - cum; // ties to take at the final threshold value
    }
  }
}

// ---------------------------------------------------------------------------
// Kernel 7: per-block partial of sum(v : v > threshold) (WMMA wave reduce).
// ---------------------------------------------------------------------------
__global__ void k_topksum(const float* __restrict__ conf_neg,
                          const unsigned* __restrict__ prefix,
                          const int* __restrict__ npos,
                          float* __restrict__ topkP) {
  __shared__ float sT[TPB / 32];
  const int b = blockIdx.y;
  const int p = blockIdx.x * TPB + threadIdx.x;
  const float v = conf_neg[b * P_ + p];
  const bool take = (npos[b] > 0) && (__float_as_uint(v) > prefix[b]);
  const float ws = wave_sum_wmma(take ? v : 0.0f);
  const int wave = threadIdx.x >> 5, lane = threadIdx.x & 31;
  if (lane == 0) sT[wave] = ws;
  __syncthreads();
  if (threadIdx.x == 0) {
    float tt = 0.0f;
    for (int i = 0; i < TPB / 32; ++i) tt += sT[i];
    topkP[b * CHUNKS + blockIdx.x] = tt;
  }
}

// ---------------------------------------------------------------------------
// Kernel 8: deterministic final combine.
// ---------------------------------------------------------------------------
__global__ void k_final(const float* __restrict__ locP,
                        const float* __restrict__ confP,
                        const float* __restrict__ topkP,
                        const int* __restrict__ npos,
                        const unsigned* __restrict__ prefix,
                        const unsigned* __restrict__ krem,
                        float* __restrict__ out) {
  __shared__ float sl[TPB], sc2[TPB], st[TPB];
  const int t = threadIdx.x;
  float a = 0.0f, c = 0.0f, tp = 0.0f;
  for (int i = t; i < NB3; i += TPB) { a += locP[i]; c += confP[i]; tp += topkP[i]; }
  sl[t] = a; sc2[t] = c; st[t] = tp;
  __syncthreads();
  for (int off = TPB / 2; off > 0; off >>= 1) {
    if (t < off) { sl[t] += sl[t + off]; sc2[t] += sc2[t + off]; st[t] += st[t + off]; }
    __syncthreads();
  }
  if (t == 0) {
    float npos_total = 0.0f, tie = 0.0f;
    for (int b = 0; b < B_; ++b) {
      npos_total += (float)npos[b];
      if (npos[b] > 0) tie += (float)krem[b] * __uint_as_float(prefix[b]);
    }
    const float den = fmaxf(npos_total, 1.0f);
    const float conf_loss = (st[0] + tie + sc2[0]) / den;
    const float loc_loss = sl[0] / den;
    out[0] = conf_loss + 0.5f * loc_loss; // ALPHA = 0.5
  }
}

// ---------------------------------------------------------------------------
extern "C" void kernel_launch(void* const* d_in, const int* in_sizes, int n_in,
                              void* d_out, int out_size, void* d_ws, size_t ws_size,
                              hipStream_t stream) {
  const float* odm_locs   = (const float*)d_in[0]; // [B,P,4]
  const float* odm_scores = (const float*)d_in[1]; // [B,P,4]
  const float* priors     = (const float*)d_in[2]; // [P,4]
  const float* boxes      = (const float*)d_in[3]; // [B,16,4]
  const int*   labels     = (const int*)d_in[4];   // [B,16]
  float* out = (float*)d_out;

  // workspace carve-up (~16.5 MB total)
  char* w = (char*)d_ws;
  size_t off = 0;
  auto carve = [&](size_t bytes) -> char* {
    char* pp = w + off;
    off = (off + bytes + 255) & ~((size_t)255);
    return pp;
  };
  unsigned* matchinfo = (unsigned*)carve((size_t)B_ * P_ * 4);
  float*    conf_neg  = (float*)carve((size_t)B_ * P_ * 4);
  float* locP  = (float*)carve((size_t)NB3 * 4);
  float* confP = (float*)carve((size_t)NB3 * 4);
  int*   nposP = (int*)carve((size_t)NB3 * 4);
  float* topkP = (float*)carve((size_t)NB3 * 4);
  unsigned long long* objbest = (unsigned long long*)carve((size_t)B_ * NOBJ * 8);
  int*      npos   = (int*)carve((size_t)B_ * 4);
  unsigned* prefix = (unsigned*)carve((size_t)B_ * 4);
  unsigned* krem   = (unsigned*)carve((size_t)B_ * 4);
  unsigned* bins   = (unsigned*)carve((size_t)B_ * 256 * 4);

  hipMemsetAsync(objbest, 0, (size_t)B_ * NOBJ * 8, stream);

  const dim3 gridBP(CHUNKS, B_);
  k_match<<<gridBP, TPB, 0, stream>>>(priors, boxes, matchinfo, objbest);
  k_force<<<B_, 32, 0, stream>>>(objbest, matchinfo);
  k_perprior<<<gridBP, TPB, 0, stream>>>(odm_locs, odm_scores, priors, boxes,
                                         labels, matchinfo, conf_neg,
                                         locP, confP, nposP);
  k_initrows<<<B_, TPB, 0, stream>>>(nposP, npos, prefix, krem, bins);
  for (int pass = 0; pass < 4; ++pass) {
    k_hist<<<gridBP, TPB, 0, stream>>>(conf_neg, prefix, krem, bins, pass);
    k_scan<<<B_, 256, 0, stream>>>(bins, prefix, krem, pass);
  }
  k_topksum<<<gridBP, TPB, 0, stream>>>(conf_neg, prefix, npos, topkP);
  k_final<<<1, TPB, 0, stream>>>(locP, confP, topkP, npos, prefix, krem, out);
}